// CombinedLoss_10479720202819
// MI455X (gfx1250) — compile-verified
//
#include <hip/hip_runtime.h>
#include <hip/hip_bf16.h>

// CombinedLoss (0.5*dice + 0.5*CE) over [8,23,512,512] f32 logits, int labels.
// Memory-bound: ~210 MB single pass => ~9us floor @ 23.3 TB/s on MI455X.
// Design: lane = pixel (coalesced per-class loads), 70 f32 accumulators held in
// VGPRs across the batch loop, wave-level reduction via V_WMMA_F32_16X16X4_F32
// (B = all-ones => B layout irrelevant, exact f32) + ds_swizzle SWAPX16 fold,
// per-wave LDS slots (plain ds_store, no divergence), global_atomic_add_f32
// grid merge, scalar finalize kernel.

#define NUM_CLASSES 23
#define HW (512 * 512)              // 262144 = 1<<18
#define BATCH 8
#define NPIX (BATCH * HW)           // 2097152
#define SMOOTH_F 1e-5f
#define WEIGHT_F 0.5f
#define NACC (3 * NUM_CLASSES + 1)  // inter[23] | sum_p[23] | counts[23] | ce
#define NWAVES 8                    // 256 threads / wave32

typedef __attribute__((ext_vector_type(2))) float v2f;
typedef __attribute__((ext_vector_type(8))) float v8f;

// Exact f32 sum of x across the 32 lanes of a wave using one WMMA.
// A(16x4 f32) documented layout: lane l, reg r -> A[l&15][2*(l>>4)+r].
// a = {x, 0} => rowsum[m] = x[m] + x[m+16] (B = all-ones, layout-irrelevant).
// D(16x16 f32): lane l VGPR v -> row v+8*(l>>4) (cols identical since B==1).
// Sum of the 8 D regs per lane = half-sum; one ds_swizzle SWAPX16 (lane^16)
// fold completes the 32-lane sum in every lane.
__device__ __forceinline__ float wave_sum_wmma(float x) {
    v2f a; a[0] = x;    a[1] = 0.0f;
    v2f b; b[0] = 1.0f; b[1] = 1.0f;
    v8f c = {};
    c = __builtin_amdgcn_wmma_f32_16x16x4_f32(
        /*neg_a=*/false, a, /*neg_b=*/false, b,
        /*c_mod=*/(short)0, c, /*reuse_a=*/false, /*reuse_b=*/false);
    float s = ((c[0] + c[1]) + (c[2] + c[3])) + ((c[4] + c[5]) + (c[6] + c[7]));
    // SWAPX16: group-of-32 swizzle, xor_mask=0x10, or=0, and_mask=0x1f
    s += __int_as_float(__builtin_amdgcn_ds_swizzle(__float_as_int(s), 0x401F));
    return s;
}

__global__ void combined_loss_zero(float* __restrict__ gacc) {
    if (threadIdx.x < NACC) gacc[threadIdx.x] = 0.0f;
}

__global__ void __launch_bounds__(256) combined_loss_main(
    const float* __restrict__ pred,   // [8,23,512,512]
    const int*   __restrict__ tgt,    // [8,1,512,512] int32 labels
    float*       __restrict__ gacc) { // [NACC]
    __shared__ float sacc[NWAVES][NACC];

    float inter[NUM_CLASSES], sump[NUM_CLASSES], cnt[NUM_CLASSES];
#pragma unroll
    for (int c = 0; c < NUM_CLASSES; ++c) { inter[c] = 0.0f; sump[c] = 0.0f; cnt[c] = 0.0f; }
    float ce = 0.0f;

    // Exactly HW threads total: thread 'hw' handles pixel (b, hw) for b=0..7.
    const unsigned hw = blockIdx.x * blockDim.x + threadIdx.x;  // < 2^18

#pragma unroll 1
    for (int b = 0; b < BATCH; ++b) {
        const float* pb = pred + ((size_t)b * NUM_CLASSES) * (size_t)HW + hw;
        const int t = tgt[(size_t)b * HW + hw];
        if (b + 1 < BATCH) {
            __builtin_prefetch(&tgt[(size_t)(b + 1) * HW + hw], 0, 0);  // global_prefetch_b8
        }

        float v[NUM_CLASSES], e[NUM_CLASSES];
        float m = -3.402823466e38f;
#pragma unroll
        for (int c = 0; c < NUM_CLASSES; ++c) {
            v[c] = pb[(size_t)c * HW];   // coalesced: consecutive lanes -> consecutive addrs
            m = fmaxf(m, v[c]);
        }
        float s = 0.0f;
#pragma unroll
        for (int c = 0; c < NUM_CLASSES; ++c) {
            e[c] = __expf(v[c] - m);
            s += e[c];
        }
        const float inv = 1.0f / s;
        const float lse = m + __logf(s);

        float vt = 0.0f;
#pragma unroll
        for (int c = 0; c < NUM_CLASSES; ++c) {
            const float pc = e[c] * inv;          // softmax prob
            sump[c] += pc;
            const bool hit = (c == t);            // all register-indexed at compile time
            inter[c] += hit ? pc : 0.0f;
            cnt[c]   += hit ? 1.0f : 0.0f;
            vt        = hit ? v[c] : vt;
        }
        ce += (lse - vt);                         // -log p_t
    }

    // Wave reduction via WMMA. EXEC is all-ones throughout (no divergence):
    // after the fold every lane holds the identical wave total, so all lanes
    // store the same value to the same per-wave LDS slot (benign, plain
    // ds_store_b32 — avoids atomics and exec manipulation entirely).
    const unsigned wid = threadIdx.x >> 5;
#pragma unroll
    for (int c = 0; c < NUM_CLASSES; ++c)
        sacc[wid][c] = wave_sum_wmma(inter[c]);
#pragma unroll
    for (int c = 0; c < NUM_CLASSES; ++c)
        sacc[wid][NUM_CLASSES + c] = wave_sum_wmma(sump[c]);
#pragma unroll
    for (int c = 0; c < NUM_CLASSES; ++c)
        sacc[wid][2 * NUM_CLASSES + c] = wave_sum_wmma(cnt[c]);
    sacc[wid][3 * NUM_CLASSES] = wave_sum_wmma(ce);
    __syncthreads();

    // Cross-wave merge + grid merge: divergent addresses -> plain
    // global_atomic_add_f32 per accumulator slot (no atomic-optimizer loops).
    if (threadIdx.x < NACC) {
        float t = 0.0f;
#pragma unroll
        for (int w = 0; w < NWAVES; ++w) t += sacc[w][threadIdx.x];
        atomicAdd(&gacc[threadIdx.x], t);
    }
}

__global__ void combined_loss_final(const float* __restrict__ gacc,
                                    float* __restrict__ out) {
    if (threadIdx.x == 0 && blockIdx.x == 0) {
        float dl = 0.0f;
        for (int c = 0; c < NUM_CLASSES; ++c) {
            const float it = gacc[c];
            const float sp = gacc[NUM_CLASSES + c];
            const float cn = gacc[2 * NUM_CLASSES + c];
            const float dice = (2.0f * it + SMOOTH_F) / (sp + cn + SMOOTH_F);
            dl += 1.0f - dice;
        }
        dl *= (1.0f / (float)NUM_CLASSES);
        const float cem = gacc[3 * NUM_CLASSES] / (float)NPIX;
        out[0] = WEIGHT_F * dl + (1.0f - WEIGHT_F) * cem;
    }
}

extern "C" void kernel_launch(void* const* d_in, const int* in_sizes, int n_in,
                              void* d_out, int out_size, void* d_ws, size_t ws_size,
                              hipStream_t stream) {
    const float* pred = (const float*)d_in[0];
    const int*   tgt  = (const int*)d_in[1];   // integer input -> const int*
    float* gacc = (float*)d_ws;                // 70 floats of scratch
    float* out  = (float*)d_out;

    hipLaunchKernelGGL(combined_loss_zero, dim3(1), dim3(128), 0, stream, gacc);
    hipLaunchKernelGGL(combined_loss_main, dim3(HW / 256), dim3(256), 0, stream,
                       pred, tgt, gacc);
    hipLaunchKernelGGL(combined_loss_final, dim3(1), dim3(32), 0, stream, gacc, out);
}